// Sampler_2886218023315
// MI455X (gfx1250) — compile-verified
//
#include <hip/hip_runtime.h>
#include <hip/hip_bf16.h>
#include <cstdint>
#include <cstddef>

#define BATCH   256
#define VOCAB   128000
#define SPLIT   5
#define SEG     (VOCAB / SPLIT)        // 25600 floats per segment
#define TPB     256                    // 8 waves (wave32)
#define TILE_F4 TPB                    // 256 float4 = 1024 floats per tile
#define TILES   (SEG / (TILE_F4 * 4))  // 25 tiles per segment
#define DEPTH   4                      // LDS stage buffers; 3 async loads in flight
#define STEADY  (TILES - (DEPTH - 1))  // 22 steady-state tiles
#define EPSF    1e-10f

// ---- CDNA5 async global->LDS copy (per-lane 16B; 512B per wave) ---------
__device__ __forceinline__ void async_load_f4(uint32_t lds_off, const float4* gsrc) {
    // GLOBAL_LOAD_ASYNC_TO_LDS_B128: vdst = LDS byte address, vaddr = 64-bit global addr
    asm volatile("global_load_async_to_lds_b128 %0, %1, off"
                 :: "v"(lds_off), "v"(gsrc)
                 : "memory");
}

// ---- Kernel 1: g[v] = log(E[v] + eps); 0.5 MB, stays L2-resident --------
__global__ void prep_log_noise(const float* __restrict__ expn, float* __restrict__ g) {
    int i = blockIdx.x * blockDim.x + threadIdx.x;
    if (i < VOCAB) g[i] = logf(expn[i] + EPSF);
}

// ---- per-tile consume, unswitched on block-uniform `greedy` -------------
template <bool GREEDY>
__device__ __forceinline__ void consume_tile(
    const float4 (*stage)[TPB], const float4* __restrict__ gseg,
    int it, int tid, int idx_base, float invT, float& bestV, int& bestI)
{
    const int fi = it * TILE_F4 + tid;                 // float4 index in segment
    float4 x = stage[it & (DEPTH - 1)][tid];           // ds_load_b128 (own data; no barrier)
    float s0, s1, s2, s3;
    if (GREEDY) {
        s0 = x.x; s1 = x.y; s2 = x.z; s3 = x.w;
    } else {
        float4 gv = gseg[fi];                          // L2-resident, shared across rows
        s0 = __builtin_fmaf(x.x, invT, -gv.x);
        s1 = __builtin_fmaf(x.y, invT, -gv.y);
        s2 = __builtin_fmaf(x.z, invT, -gv.z);
        s3 = __builtin_fmaf(x.w, invT, -gv.w);
    }
    const int e = idx_base + fi * 4;
    if (s0 > bestV) { bestV = s0; bestI = e;     }     // strict > keeps first index
    if (s1 > bestV) { bestV = s1; bestI = e + 1; }
    if (s2 > bestV) { bestV = s2; bestI = e + 2; }
    if (s3 > bestV) { bestV = s3; bestI = e + 3; }
}

template <bool GREEDY>
__device__ __forceinline__ void stream_segment(
    const float4 (*stage)[TPB], const float4* __restrict__ rowp,
    const float4* __restrict__ gseg, int tid, int idx_base, float invT,
    float& bestV, int& bestI)
{
    // Prologue: 3 tiles in flight.
    #pragma unroll
    for (int p = 0; p < DEPTH - 1; ++p) {
        uint32_t loff = (uint32_t)(size_t)&stage[p][tid];  // generic ptr low 32 = DS offset
        async_load_f4(loff, rowp + p * TILE_F4 + tid);
    }
    // Steady state: literal wait, consume, refill.
    #pragma unroll 2
    for (int it = 0; it < STEADY; ++it) {
        asm volatile("s_wait_asynccnt 0x2" ::: "memory");  // tile `it` complete (in-order)
        consume_tile<GREEDY>(stage, gseg, it, tid, idx_base, invT, bestV, bestI);
        const int nt = it + (DEPTH - 1);
        uint32_t loff = (uint32_t)(size_t)&stage[nt & (DEPTH - 1)][tid];
        async_load_f4(loff, rowp + nt * TILE_F4 + tid);
    }
    // Epilogue: drain last 3 tiles with literal waits.
    asm volatile("s_wait_asynccnt 0x2" ::: "memory");
    consume_tile<GREEDY>(stage, gseg, STEADY + 0, tid, idx_base, invT, bestV, bestI);
    asm volatile("s_wait_asynccnt 0x1" ::: "memory");
    consume_tile<GREEDY>(stage, gseg, STEADY + 1, tid, idx_base, invT, bestV, bestI);
    asm volatile("s_wait_asynccnt 0x0" ::: "memory");
    consume_tile<GREEDY>(stage, gseg, STEADY + 2, tid, idx_base, invT, bestV, bestI);
}

// ---- Kernel 2: streaming fused score + partial argmax -------------------
__global__ void __launch_bounds__(TPB) argmax_partial(
    const float* __restrict__ logits,
    const float* __restrict__ temps,
    const float* __restrict__ g,
    float* __restrict__ pv,
    int*   __restrict__ pi)
{
    __shared__ float4 stage[DEPTH][TPB];   // 16 KB async staging
    __shared__ float  rv[TPB];
    __shared__ int    ri[TPB];

    const int tid = threadIdx.x;
    const int row = blockIdx.x / SPLIT;
    const int seg = blockIdx.x % SPLIT;

    const float t      = temps[row];
    const bool  greedy = (t == 0.0f);
    const float invT   = greedy ? 1.0f : (1.0f / t);

    const float4* rowp = (const float4*)(logits + (size_t)row * VOCAB) + seg * (SEG / 4);
    const float4* gseg = (const float4*)g + seg * (SEG / 4);
    const int idx_base = seg * SEG;

    float bestV = -__builtin_huge_valf();
    int   bestI = 0;

    if (greedy) stream_segment<true >(stage, rowp, gseg, tid, idx_base, invT, bestV, bestI);
    else        stream_segment<false>(stage, rowp, gseg, tid, idx_base, invT, bestV, bestI);

    // Block reduction with first-index tie-break.
    rv[tid] = bestV;
    ri[tid] = bestI;
    __syncthreads();
    #pragma unroll
    for (int s = TPB / 2; s > 0; s >>= 1) {
        if (tid < s) {
            float v2 = rv[tid + s];
            int   i2 = ri[tid + s];
            if (v2 > rv[tid] || (v2 == rv[tid] && i2 < ri[tid])) {
                rv[tid] = v2; ri[tid] = i2;
            }
        }
        __syncthreads();
    }
    if (tid == 0) {
        pv[blockIdx.x] = rv[0];
        pi[blockIdx.x] = ri[0];
    }
}

// ---- Kernel 3: fold 5 ordered segment partials per row ------------------
__global__ void finalize(const float* __restrict__ pv, const int* __restrict__ pi,
                         int* __restrict__ out)
{
    const int r = threadIdx.x;   // one thread per row, 256 threads
    float bv = -__builtin_huge_valf();
    int   bi = 0;
    #pragma unroll
    for (int s = 0; s < SPLIT; ++s) {     // ascending index ranges: strict > keeps lowest idx
        float v = pv[r * SPLIT + s];
        int   i = pi[r * SPLIT + s];
        if (v > bv) { bv = v; bi = i; }
    }
    out[r] = bi;
}

extern "C" void kernel_launch(void* const* d_in, const int* in_sizes, int n_in,
                              void* d_out, int out_size, void* d_ws, size_t ws_size,
                              hipStream_t stream) {
    const float* logits = (const float*)d_in[0];   // [256, 128000] f32
    const float* temps  = (const float*)d_in[1];   // [256] f32
    const float* expn   = (const float*)d_in[2];   // [128000] f32

    float* g  = (float*)d_ws;                      // [VOCAB] log-noise
    float* pv = g + VOCAB;                         // [BATCH*SPLIT] partial vals
    int*   pi = (int*)(pv + BATCH * SPLIT);        // [BATCH*SPLIT] partial idx
    int*   out = (int*)d_out;                      // [256] int32 tokens

    prep_log_noise<<<(VOCAB + TPB - 1) / TPB, TPB, 0, stream>>>(expn, g);
    argmax_partial<<<BATCH * SPLIT, TPB, 0, stream>>>(logits, temps, g, pv, pi);
    finalize<<<1, BATCH, 0, stream>>>(pv, pi, out);
}